// AttentionModule_40450001994093
// MI455X (gfx1250) — compile-verified
//
#include <hip/hip_runtime.h>
#include <math.h>

constexpr int kT    = 524288;  // tokens
constexpr int kD    = 256;     // embedding dim
constexpr int kN    = 8192;    // segments

typedef __attribute__((ext_vector_type(2))) float v2f;
typedef __attribute__((ext_vector_type(8))) float v8f;

__device__ __forceinline__ float fast_tanh(float x) {
  float r;
  asm("v_tanh_f32 %0, %1" : "=v"(r) : "v"(x));  // CDNA5 hardware transcendental
  return r;
}

// ---------------- zero workspace (sums, rep, counts) ----------------
__global__ void k_zero(float* __restrict__ p, int n) {
  int i = blockIdx.x * blockDim.x + threadIdx.x;
  int stride = gridDim.x * blockDim.x;
  for (; i < n; i += stride) p[i] = 0.0f;
}

// ---------------- pass A: per-segment sums of embedding + counts ----------------
// obj_to_img is sorted: accumulate runs in registers, flush with one atomic per run.
// Block = 256 threads (one per column d), 256 tokens per block.
__global__ void k_segsum(const float* __restrict__ emb, const int* __restrict__ seg,
                         float* __restrict__ sums, float* __restrict__ counts) {
  __shared__ int sseg[256];
  const int d  = threadIdx.x;
  const int t0 = blockIdx.x * 256;
  sseg[d] = seg[t0 + d];
  __syncthreads();

  int   cur = sseg[0];
  float acc = 0.0f, cnt = 0.0f;
  for (int i = 0; i < 256; ++i) {
    const int   s = sseg[i];                         // uniform per block -> no divergence
    const float v = emb[(size_t)(t0 + i) * kD + d];  // coalesced 1KB row
    if (s != cur) {
      atomicAdd(&sums[(size_t)cur * kD + d], acc);
      if (d == 0) atomicAdd(&counts[cur], cnt);
      acc = 0.0f; cnt = 0.0f; cur = s;
    }
    acc += v; cnt += 1.0f;
  }
  atomicAdd(&sums[(size_t)cur * kD + d], acc);
  if (d == 0) atomicAdd(&counts[cur], cnt);
}

// ---------------- GEMM [N,256]x[256,256] via V_WMMA_F32_16X16X4_F32, fused tanh(./count) ----
// One 16x16 C tile per wave, K-loop in steps of 4. 8 waves per 256-thread block.
__global__ void k_gemm_tanh(const float* __restrict__ sums, const float* __restrict__ W,
                            const float* __restrict__ counts, float* __restrict__ tg) {
  const int lane = threadIdx.x & 31;
  const int wid  = threadIdx.x >> 5;
  const int tile = blockIdx.x * 8 + wid;   // 8192 tiles total
  const int tm   = tile >> 4;              // kD/16 == 16 tiles along N-dim of C
  const int tn   = tile & 15;
  const int m0   = tm * 16;
  const int n0   = tn * 16;
  const int l16  = lane & 15;
  const int koff = (lane < 16) ? 0 : 2;    // ISA 32-bit A 16x4 layout

  v8f c = {};
  for (int k0 = 0; k0 < kD; k0 += 4) {
    const int k = k0 + koff;
    v2f a, b;
    // A (16x4): lane holds row m = lane%16; VGPR0 = K (0|2), VGPR1 = K+1 (1|3)
    a.x = sums[(size_t)(m0 + l16) * kD + k];
    a.y = sums[(size_t)(m0 + l16) * kD + k + 1];
    // B (4x16): row K striped across lanes within a VGPR, same half-split as A
    b.x = W[(size_t)k * kD + n0 + l16];
    b.y = W[(size_t)(k + 1) * kD + n0 + l16];
    c = __builtin_amdgcn_wmma_f32_16x16x4_f32(
        /*neg_a=*/false, a, /*neg_b=*/false, b,
        /*c_mod=*/(short)0, c, /*reuse_a=*/false, /*reuse_b=*/false);
  }

  // C layout: VGPR r -> row m0 + r + (lane<16 ? 0 : 8), col n0 + lane%16
  const int rbase = m0 + ((lane < 16) ? 0 : 8);
  const int col   = n0 + l16;
#pragma unroll
  for (int r = 0; r < 8; ++r) {
    const int   row = rbase + r;
    const float inv = __builtin_amdgcn_rcpf(fmaxf(counts[row], 1.0f));  // counts exact ints
    tg[(size_t)row * kD + col] = fast_tanh(c[r] * inv);
  }
}

// ---------------- pass B: fused score + weighted segment sum ----------------
// One wave per 64-token chunk; lane owns 8 columns. Per token: dot(e, tg[seg])
// via wave32 shfl_xor reduction, sigmoid, then register-accumulated rep run.
__global__ void k_score_rep(const float* __restrict__ emb, const int* __restrict__ seg,
                            const float* __restrict__ tg, float* __restrict__ rep) {
  const int gtid = blockIdx.x * blockDim.x + threadIdx.x;
  const int wave = gtid >> 5;
  const int lane = threadIdx.x & 31;
  const int t0   = wave * 64;
  const int cb   = lane * 8;

  float g[8], acc[8];
  int cur = seg[t0];
#pragma unroll
  for (int j = 0; j < 8; ++j) { g[j] = tg[(size_t)cur * kD + cb + j]; acc[j] = 0.0f; }

  for (int i = 0; i < 64; ++i) {
    const int t = t0 + i;
    const int s = seg[t];
    if (s != cur) {  // flush run, reload tg row
#pragma unroll
      for (int j = 0; j < 8; ++j) {
        atomicAdd(&rep[(size_t)cur * kD + cb + j], acc[j]);
        acc[j] = 0.0f;
        g[j]   = tg[(size_t)s * kD + cb + j];
      }
      cur = s;
    }
    float e[8];
    const float* er = emb + (size_t)t * kD + cb;
    // deepen the memory pipeline: prefetch 16 tokens ahead (bounded)
    const int tp = (t + 16 < kT) ? (t + 16) : (kT - 1);
    __builtin_prefetch(emb + (size_t)tp * kD + cb, 0, 0);
#pragma unroll
    for (int j = 0; j < 8; ++j) e[j] = er[j];   // 2x b128 loads, 1KB/wave coalesced
    float p = 0.0f;
#pragma unroll
    for (int j = 0; j < 8; ++j) p += e[j] * g[j];
#pragma unroll
    for (int off = 16; off >= 1; off >>= 1) p += __shfl_xor(p, off, 32);
    const float score = __builtin_amdgcn_rcpf(1.0f + __expf(-p));  // sigmoid, no IEEE div
#pragma unroll
    for (int j = 0; j < 8; ++j) acc[j] += score * e[j];
  }
#pragma unroll
  for (int j = 0; j < 8; ++j) atomicAdd(&rep[(size_t)cur * kD + cb + j], acc[j]);
}

// ---------------- pass C: out[t] = rep[seg[t]] (broadcast) ----------------
// 64 tokens per block; float4 stores, rep rows are L2-hot (8 MB total).
__global__ void k_bcast(const float* __restrict__ rep, const int* __restrict__ seg,
                        float* __restrict__ out) {
  __shared__ int sseg[64];
  const int tid = threadIdx.x;
  const int t0  = blockIdx.x * 64;
  if (tid < 64) sseg[tid] = seg[t0 + tid];
  __syncthreads();

  const int sub = tid >> 6;        // 4 tokens in flight per iteration
  const int c4  = (tid & 63) * 4;  // float4 column
  for (int i = 0; i < 64; i += 4) {
    const int t = t0 + i + sub;
    const int s = sseg[i + sub];
    const float4 v = *(const float4*)&rep[(size_t)s * kD + c4];
    *(float4*)&out[(size_t)t * kD + c4] = v;
  }
}

extern "C" void kernel_launch(void* const* d_in, const int* in_sizes, int n_in,
                              void* d_out, int out_size, void* d_ws, size_t ws_size,
                              hipStream_t stream) {
  const float* emb = (const float*)d_in[0];
  const float* W   = (const float*)d_in[1];
  const int*   seg = (const int*)d_in[2];
  // d_in[3] = num_segments (compile-time kN)
  float* out = (float*)d_out;

  float* ws = (float*)d_ws;
  const size_t ND = (size_t)kN * kD;
  float* sums   = ws;                  // [N, D]  8 MB
  float* rep    = ws + ND;             // [N, D]  8 MB
  float* counts = ws + 2 * ND;         // [N]     32 KB
  float* tg     = ws + 2 * ND + kN;    // [N, D]  8 MB (fully overwritten, no zeroing)

  const int nzero = (int)(2 * ND + kN);
  k_zero<<<2048, 256, 0, stream>>>(ws, nzero);
  k_segsum<<<kT / 256, 256, 0, stream>>>(emb, seg, sums, counts);
  k_gemm_tanh<<<(kN / 16) * (kD / 16) / 8, 256, 0, stream>>>(sums, W, counts, tg);
  k_score_rep<<<(kT / 64) / 8, 256, 0, stream>>>(emb, seg, tg, rep);
  k_bcast<<<kT / 64, 256, 0, stream>>>(rep, seg, out);
}